// ExpandFrame_79499844649648
// MI455X (gfx1250) — compile-verified
//
#include <hip/hip_runtime.h>
#include <hip/hip_bf16.h>

typedef float v2f __attribute__((ext_vector_type(2)));
typedef float v8f __attribute__((ext_vector_type(8)));

#define B_ 32
#define N_ 1024
#define D_ 256

// ---------------------------------------------------------------------------
// k_prep: per-batch  S[b] = sum_n floor(dur+1.5);  center[b,n] = S - 0.5*d;
//         also zero-initialize the esum accumulator region of the workspace.
// ---------------------------------------------------------------------------
__global__ void k_prep(const float* __restrict__ dur,
                       float* __restrict__ center,
                       float* __restrict__ esum) {
  const int b   = blockIdx.x;
  const int tid = threadIdx.x;                    // 256 threads
  __shared__ float red[256];

  // zero esum accumulators (atomics target in k_esum)
  esum[b * D_ + tid] = 0.0f;

  float s = 0.0f;
  for (int n = tid; n < N_; n += 256)
    s += floorf(dur[b * N_ + n] + 1.5f);
  red[tid] = s;
  __syncthreads();
  for (int off = 128; off > 0; off >>= 1) {
    if (tid < off) red[tid] += red[tid + off];
    __syncthreads();
  }
  const float S = red[0];

  for (int n = tid; n < N_; n += 256) {
    float d = floorf(dur[b * N_ + n] + 1.5f);
    center[b * N_ + n] = S - 0.5f * d;
  }
}

// ---------------------------------------------------------------------------
// k_esum: e_sum[b, d0:d0+16] = sum over N rows of enc[b, :, d0:d0+16]
// Implemented as ones(16x4) x B(4x16) WMMA accumulation: with A all-ones,
// C[m,n] = sum_k B[k,n] independent of the K->slot mapping, so each lane
// loads a unique (row, col) pair covering rows k..k+3, cols d0..d0+15.
// One wave per block (EXEC all ones, as WMMA requires). Partial sums over an
// N-slice are combined with global_atomic_add_f32.
// ---------------------------------------------------------------------------
__global__ void k_esum(const float* __restrict__ enc,
                       float* __restrict__ esum) {
  const int b    = blockIdx.x;        // batch
  const int dc   = blockIdx.y;        // d-chunk 0..15
  const int ns   = blockIdx.z;        // n-slice 0..3
  const int lane = threadIdx.x;       // 0..31 (wave32)
  const int d0   = dc * 16;
  const int col  = lane & 15;
  const int half = lane >> 4;         // 0: rows k,k+1   1: rows k+2,k+3

  const float* base = enc + (size_t)b * N_ * D_ + d0 + col;

  v8f c = {0.f, 0.f, 0.f, 0.f, 0.f, 0.f, 0.f, 0.f};
  v2f a;
  a.x = 1.0f;
  a.y = 1.0f;

  const int k0 = ns * (N_ / 4);
  const int k1 = k0 + (N_ / 4);
  for (int k = k0; k < k1; k += 4) {
    const float* p = base + (size_t)(k + 2 * half) * D_;
    v2f bv;
    bv.x = p[0];
    bv.y = p[D_];
    // D = A(ones 16x4, f32) * B(4x16, f32) + C   -> v_wmma_f32_16x16x4_f32
    c = __builtin_amdgcn_wmma_f32_16x16x4_f32(
        /*neg_a=*/false, a, /*neg_b=*/false, bv,
        /*c_mod=*/(short)0, c, /*reuse_a=*/false, /*reuse_b=*/false);
  }

  // Row 0 of C lives in VGPR0 lanes 0..15 (N = lane). All rows identical.
  if (lane < 16) atomicAdd(&esum[b * D_ + d0 + lane], c[0]);
}

// ---------------------------------------------------------------------------
// k_wlast: denom[b,t] = sum_n exp(-0.1*(t-center)^2);
//          w_last[b,t] = exp(-0.1*(t-center[b,N-1])^2) / denom
// One thread per (b,t); center rows (4KB/batch) are L2-resident and reused
// by 2048 threads. Inner loop is float4-vectorized: 4 v_exp_f32 per load.
// ---------------------------------------------------------------------------
__global__ void k_wlast(const float* __restrict__ center,
                        float* __restrict__ wlast, int T, int nbt) {
  const int idx = blockIdx.x * blockDim.x + threadIdx.x;
  if (idx >= nbt) return;
  const int b = idx / T;
  const int t = idx - b * T;
  const float tf = (float)t;

  const float4* c4 = (const float4*)(center + b * N_);
  float acc = 0.0f;
#pragma unroll 4
  for (int i = 0; i < N_ / 4; ++i) {
    float4 cv = c4[i];
    float x0 = tf - cv.x;
    float x1 = tf - cv.y;
    float x2 = tf - cv.z;
    float x3 = tf - cv.w;
    acc += __expf(-0.1f * x0 * x0);
    acc += __expf(-0.1f * x1 * x1);
    acc += __expf(-0.1f * x2 * x2);
    acc += __expf(-0.1f * x3 * x3);
  }
  const float cl = center[b * N_ + N_ - 1];
  const float xl = tf - cl;
  wlast[idx] = __expf(-0.1f * xl * xl) / acc;
}

// ---------------------------------------------------------------------------
// k_out: out[b,t,:] = w_last[b,t] * e_sum[b,:]   (64 MB of float4 stores —
// the bandwidth-dominant kernel; w_last and e_sum hit L2.)
// ---------------------------------------------------------------------------
__global__ void k_out(const float* __restrict__ wlast,
                      const float* __restrict__ esum,
                      float4* __restrict__ out, int T, int total4) {
  const int i = blockIdx.x * blockDim.x + threadIdx.x;
  if (i >= total4) return;
  const int d4 = i & (D_ / 4 - 1);   // 0..63
  const int bt = i >> 6;             // b*T + t
  const int b  = bt / T;
  const float w = wlast[bt];
  const float4 e = ((const float4*)esum)[b * (D_ / 4) + d4];
  out[i] = make_float4(w * e.x, w * e.y, w * e.z, w * e.w);
}

extern "C" void kernel_launch(void* const* d_in, const int* in_sizes, int n_in,
                              void* d_out, int out_size, void* d_ws, size_t ws_size,
                              hipStream_t stream) {
  const float* enc = (const float*)d_in[0];   // [B, N, D] f32
  const float* dur = (const float*)d_in[1];   // [B, N, 1] f32
  float* out = (float*)d_out;                 // [B, T, D] f32

  const int T = out_size / (B_ * D_);         // 2048 for these inputs

  // workspace layout (floats): center [B*N] | esum [B*D] | wlast [B*T]
  float* ws     = (float*)d_ws;
  float* center = ws;
  float* esum   = center + B_ * N_;
  float* wlast  = esum + B_ * D_;

  k_prep<<<B_, 256, 0, stream>>>(dur, center, esum);

  dim3 g_esum(B_, D_ / 16, 4);
  k_esum<<<g_esum, 32, 0, stream>>>(enc, esum);

  const int nbt = B_ * T;
  k_wlast<<<(nbt + 255) / 256, 256, 0, stream>>>(center, wlast, T, nbt);

  const int total4 = out_size / 4;
  k_out<<<(total4 + 255) / 256, 256, 0, stream>>>(wlast, esum, (float4*)out, T, total4);
}